// CSA_25168508354945
// MI455X (gfx1250) — compile-verified
//
#include <hip/hip_runtime.h>

// CDNA5 (gfx1250) wave32 + V_WMMA_F32_16X16X4_F32 fused self-attention.
// Stage 1: q/k/v 1x1-conv projections via f32 WMMA into d_ws (16 MB).
// Stage 2: flash-attention style fused energy/softmax/PV, f32 WMMA throughout,
//          with global_prefetch of the next K/V tile overlapped with compute.

typedef float v2f __attribute__((ext_vector_type(2)));
typedef float v8f __attribute__((ext_vector_type(8)));

#define WMMA_F32X4(A, B, C) \
  __builtin_amdgcn_wmma_f32_16x16x4_f32(false, (A), false, (B), (short)0, (C), false, false)

namespace {
constexpr int BATCH = 4;
constexpr int CQK   = 64;
constexpr int CV    = 128;
constexpr int NSEQ  = 4096;
constexpr int MT    = 32;            // keys processed per flash iteration
// Pair-interleaved LDS layouts: [pair_row][col][2] so each WMMA operand is one b64.
// Row strides kept even so v2f (8B) LDS loads stay aligned.
constexpr int LDK2 = 2 * MT + 2;     // 66 dwords per K pair-row (32 rows)
constexpr int LDV2 = 2 * CV + 2;     // 258 dwords per V pair-row (16 rows)
constexpr int LDP2 = 2 * 16 + 2;     // 34 dwords per P pair-row (16 rows/wave)
}

// ---------------------------------------------------------------------------
// Stage 1: Y[b,o,n] = sum_c W[o,c] * X[b,c,n]
// One wave computes a 16(o) x 64(n) tile. 4 waves/block cover 64 o-rows.
// grid = (NSEQ/64, O/64, BATCH), block = 128.
// ---------------------------------------------------------------------------
__global__ __launch_bounds__(128) void proj_wmma_kernel(
    const float* __restrict__ W, const float* __restrict__ X,
    float* __restrict__ Y, int O, int C) {
  const int lane = threadIdx.x & 31;
  const int wave = threadIdx.x >> 5;
  const int h    = lane >> 4;        // lane half (K split)
  const int l15  = lane & 15;
  const int b    = blockIdx.z;
  const int o0   = blockIdx.y * 64 + wave * 16;
  const int n0   = blockIdx.x * 64;

  const float* Xb = X + (size_t)b * C * NSEQ;
  float* Yb       = Y + (size_t)b * O * NSEQ;

  v8f acc[4];
#pragma unroll
  for (int st = 0; st < 4; ++st) { v8f z = {}; acc[st] = z; }

  const int ksteps = C >> 2;
  for (int t = 0; t < ksteps; ++t) {
    const int c0 = 4 * t + 2 * h;
    // A = W strip [16 o x 4 c]; elems = consecutive c -> one 8B load.
    v2f a = *(const v2f*)&W[(size_t)(o0 + l15) * C + c0];
#pragma unroll
    for (int st = 0; st < 4; ++st) {
      const int n = n0 + 16 * st + l15;
      v2f bop;
      bop[0] = Xb[(size_t)c0 * NSEQ + n];
      bop[1] = Xb[(size_t)(c0 + 1) * NSEQ + n];
      acc[st] = WMMA_F32X4(a, bop, acc[st]);
    }
  }
#pragma unroll
  for (int st = 0; st < 4; ++st) {
    const int n = n0 + 16 * st + l15;
#pragma unroll
    for (int j = 0; j < 8; ++j) {
      Yb[(size_t)(o0 + j + 8 * h) * NSEQ + n] = acc[st][j];
    }
  }
}

// ---------------------------------------------------------------------------
// Stage 2: fused attention.
// out[b,c,n] = gamma * ( sum_m softmax_m(q^T k)[n,m] * v[c,m] / l[n] ) + x_h[b,c,n]
// Block = 4 waves; wave w owns query rows n0 = blockIdx.x*64 + 16*w.
// grid = (NSEQ/64, BATCH).
// ---------------------------------------------------------------------------
__global__ __launch_bounds__(128) void flash_attn_f32_kernel(
    const float* __restrict__ q, const float* __restrict__ k,
    const float* __restrict__ v, const float* __restrict__ x_h,
    const float* __restrict__ gamma_p, float* __restrict__ out) {
  __shared__ float kbp[(CQK / 2) * LDK2];        // K tile, pair-interleaved on c
  __shared__ float vtp[(MT / 2) * LDV2];         // V tile, pair-interleaved on m
  __shared__ float ptb[4 * (MT / 2) * LDP2];     // per-wave P transpose buffers
  __shared__ float red[4][16];                   // per-wave row-stat exchange
  __shared__ float mrow_s[4][16];                // per-wave m_new broadcast

  const int tid  = threadIdx.x;
  const int lane = tid & 31;
  const int wave = tid >> 5;
  const int h    = lane >> 4;
  const int l15  = lane & 15;
  const int b    = blockIdx.y;
  const int n0   = blockIdx.x * 64 + wave * 16;  // this wave's 16 query rows

  const float* qb  = q + (size_t)b * CQK * NSEQ;
  const float* kb  = k + (size_t)b * CQK * NSEQ;
  const float* vb  = v + (size_t)b * CV * NSEQ;

  // Per-thread staging coordinates (also used for prefetch of next tile).
  const int kc  = tid >> 1;            // K tile: row c, half-row of m
  const int kmh = (tid & 1) * 16;
  const int vc  = tid;                 // V tile: row c, all 32 m

  // Load q^T A-operands for this wave's strip: 16 ksteps x (16n x 4c).
  v2f qa[16];
#pragma unroll
  for (int t = 0; t < 16; ++t) {
    const int c0 = 4 * t + 2 * h;
    const int n  = n0 + l15;
    qa[t][0] = qb[(size_t)c0 * NSEQ + n];
    qa[t][1] = qb[(size_t)(c0 + 1) * NSEQ + n];
  }

  v8f Oacc[8];
#pragma unroll
  for (int ct = 0; ct < 8; ++ct) { v8f z = {}; Oacc[ct] = z; }
  float m_s = -3.0e38f;  // running max for row n = n0 + l15 (dup across halves)
  float l_s = 0.0f;      // running denominator

  float* ptw = &ptb[wave * (MT / 2) * LDP2];

  for (int m0 = 0; m0 < NSEQ; m0 += MT) {
    __syncthreads();  // previous iteration's LDS reads done before overwrite

    // --- cooperative stage K tile [64 c x 32 m] -> kbp[(c>>1)][m][c&1] ---
    {
      float* dst = &kbp[(kc >> 1) * LDK2 + (kc & 1)];
#pragma unroll
      for (int i = 0; i < 4; ++i) {
        const int mb = kmh + 4 * i;
        const float4 kv = *(const float4*)&kb[(size_t)kc * NSEQ + m0 + mb];
        dst[(mb + 0) * 2] = kv.x;
        dst[(mb + 1) * 2] = kv.y;
        dst[(mb + 2) * 2] = kv.z;
        dst[(mb + 3) * 2] = kv.w;
      }
    }
    // --- cooperative stage V tile [128 c x 32 m] -> vtp[(m>>1)][c][m&1] ---
    {
#pragma unroll
      for (int i = 0; i < 8; ++i) {
        const int mb = 4 * i;
        const float4 vv = *(const float4*)&vb[(size_t)vc * NSEQ + m0 + mb];
        vtp[(2 * i + 0) * LDV2 + 2 * vc + 0] = vv.x;
        vtp[(2 * i + 0) * LDV2 + 2 * vc + 1] = vv.y;
        vtp[(2 * i + 1) * LDV2 + 2 * vc + 0] = vv.z;
        vtp[(2 * i + 1) * LDV2 + 2 * vc + 1] = vv.w;
      }
    }
    __syncthreads();

    // Prefetch next iteration's K/V tiles (global_prefetch_b8) so HBM/L2
    // latency overlaps with the 96 WMMAs below. Each thread touches exactly
    // the lines it will load next iteration; no data/ordering semantics.
    if (m0 + MT < NSEQ) {
      __builtin_prefetch(&kb[(size_t)kc * NSEQ + m0 + MT + kmh], 0, 1);
      __builtin_prefetch(&vb[(size_t)vc * NSEQ + m0 + MT], 0, 1);
    }

    // --- energy tile E[16 n x 32 m] = q^T k (f32 WMMA, K=64 channels) ---
    v8f E[2];
#pragma unroll
    for (int st = 0; st < 2; ++st) { v8f z = {}; E[st] = z; }
    for (int t = 0; t < 16; ++t) {
      const float* krow = &kbp[(2 * t + h) * LDK2];
#pragma unroll
      for (int st = 0; st < 2; ++st) {
        v2f bop = *(const v2f*)&krow[(16 * st + l15) * 2];
        E[st] = WMMA_F32X4(qa[t], bop, E[st]);
      }
    }

    // --- online softmax: row max over 32 keys ---
    v8f tmax;
#pragma unroll
    for (int j = 0; j < 8; ++j) tmax[j] = fmaxf(E[0][j], E[1][j]);
#pragma unroll
    for (int d = 1; d <= 8; d <<= 1) {
#pragma unroll
      for (int j = 0; j < 8; ++j)
        tmax[j] = fmaxf(tmax[j], __shfl_xor(tmax[j], d, 32));
    }
    if (l15 == 0) {  // lanes 0 / 16 publish rows 0-7 / 8-15
#pragma unroll
      for (int j = 0; j < 8; ++j) red[wave][j + 8 * h] = tmax[j];
    }
    const float rmax = red[wave][l15];
    const float mnew = fmaxf(m_s, rmax);
    const float scale = __expf(m_s - mnew);
    m_s = mnew;
    if (h == 0) mrow_s[wave][l15] = mnew;  // broadcast per-row m_new

    // P = exp(E - m_new) in the E (row-indexed) layout
    v8f mrow;
#pragma unroll
    for (int j = 0; j < 8; ++j) mrow[j] = mrow_s[wave][j + 8 * h];
#pragma unroll
    for (int st = 0; st < 2; ++st)
#pragma unroll
      for (int j = 0; j < 8; ++j) E[st][j] = __expf(E[st][j] - mrow[j]);

    // row sum of P
    v8f tsum;
#pragma unroll
    for (int j = 0; j < 8; ++j) tsum[j] = E[0][j] + E[1][j];
#pragma unroll
    for (int d = 1; d <= 8; d <<= 1) {
#pragma unroll
      for (int j = 0; j < 8; ++j) tsum[j] += __shfl_xor(tsum[j], d, 32);
    }
    if (l15 == 0) {
#pragma unroll
      for (int j = 0; j < 8; ++j) red[wave][j + 8 * h] = tsum[j];
    }
    l_s = l_s * scale + red[wave][l15];

    // rescale output accumulator (per-lane scalar: O's N-dim is n = n0+l15)
#pragma unroll
    for (int ct = 0; ct < 8; ++ct)
#pragma unroll
      for (int j = 0; j < 8; ++j) Oacc[ct][j] *= scale;

    // stage P into per-wave transpose buffer: ptw[(m>>1)][n][m&1]
#pragma unroll
    for (int st = 0; st < 2; ++st) {
      const int m = 16 * st + l15;
#pragma unroll
      for (int j = 0; j < 8; ++j)
        ptw[(m >> 1) * LDP2 + (j + 8 * h) * 2 + (m & 1)] = E[st][j];
    }

    // --- PV: out[c,n] += V[16c x 4m] * P^T[4m x 16n], K = 32 keys ---
    for (int t = 0; t < 8; ++t) {
      const int pairm = 2 * t + h;
      v2f bop = *(const v2f*)&ptw[pairm * LDP2 + l15 * 2];
      const float* vrow = &vtp[pairm * LDV2];
#pragma unroll
      for (int ct = 0; ct < 8; ++ct) {
        v2f aop = *(const v2f*)&vrow[(16 * ct + l15) * 2];
        Oacc[ct] = WMMA_F32X4(aop, bop, Oacc[ct]);
      }
    }
  }

  // --- epilogue: out = gamma * (O / l) + x_h, coalesced along n ---
  const float g = gamma_p[0];
  const float inv_l = 1.0f / l_s;
  const float* xhb = x_h + (size_t)b * CV * NSEQ;
  float* ob        = out + (size_t)b * CV * NSEQ;
  const int n = n0 + l15;
#pragma unroll
  for (int ct = 0; ct < 8; ++ct) {
#pragma unroll
    for (int j = 0; j < 8; ++j) {
      const int c = 16 * ct + j + 8 * h;
      const size_t idx = (size_t)c * NSEQ + n;
      ob[idx] = g * (Oacc[ct][j] * inv_l) + xhb[idx];
    }
  }
}

// ---------------------------------------------------------------------------
extern "C" void kernel_launch(void* const* d_in, const int* in_sizes, int n_in,
                              void* d_out, int out_size, void* d_ws, size_t ws_size,
                              hipStream_t stream) {
  (void)in_sizes; (void)n_in; (void)out_size; (void)ws_size;
  const float* x     = (const float*)d_in[0];  // [B, 64, N]
  const float* x_h   = (const float*)d_in[1];  // [B, 128, N]
  const float* Wq    = (const float*)d_in[2];  // [64, 64]
  const float* Wk    = (const float*)d_in[3];  // [64, 64]
  const float* Wv    = (const float*)d_in[4];  // [128, 128]
  const float* gamma = (const float*)d_in[5];  // [1]
  float* out = (float*)d_out;

  // Workspace: q (4MB) | k (4MB) | v (8MB) = 16MB total, all f32 [b][c][n].
  float* qw = (float*)d_ws;
  float* kw = qw + (size_t)BATCH * CQK * NSEQ;
  float* vw = kw + (size_t)BATCH * CQK * NSEQ;

  dim3 blk(128);
  proj_wmma_kernel<<<dim3(NSEQ / 64, CQK / 64, BATCH), blk, 0, stream>>>(Wq, x, qw, CQK, CQK);
  proj_wmma_kernel<<<dim3(NSEQ / 64, CQK / 64, BATCH), blk, 0, stream>>>(Wk, x, kw, CQK, CQK);
  proj_wmma_kernel<<<dim3(NSEQ / 64, CV / 64, BATCH), blk, 0, stream>>>(Wv, x_h, vw, CV, CV);
  flash_attn_f32_kernel<<<dim3(NSEQ / 64, BATCH), blk, 0, stream>>>(qw, kw, vw, x_h, gamma, out);
}